// VolatilityTrackingLayer_52235392254129
// MI455X (gfx1250) — compile-verified
//
#include <hip/hip_runtime.h>

// CDNA5 / gfx1250, wave32.
typedef __attribute__((ext_vector_type(2))) float v2f;
typedef __attribute__((ext_vector_type(8))) float v8f;

#define S_LEN 8192
#define H_DIM 256
#define W_WIN 64
#define LN2F  0.6931471805599453f

// Block: 512 threads = 16 wave32 waves.
// Each block handles 16 consecutive time positions of one batch row,
// producing a 16x256 output tile via 16 V_WMMA_F32_16X16X4_F32 ops
// (one 16x16 tile per wave). K is padded 3->4: A[:,3]=1.0, B[3,:]=bias,
// so the bias add is fused into the WMMA accumulate.
__global__ __launch_bounds__(512)
void VolatilityTrackingLayer_kernel(const float* __restrict__ x,
                                    const float* __restrict__ lin_w,
                                    const float* __restrict__ lin_b,
                                    float* __restrict__ out) {
    __shared__ float s_close[80];   // closes t0-64 .. t0+14 (79 used)
    __shared__ float s_r[80];       // 78 log-returns shared by the 16 windows
    __shared__ float s_A[16][4];    // [std, gk, pk, 1.0] per tile row

    const int b   = blockIdx.y;
    const int t0  = blockIdx.x * 16;
    const int tid = threadIdx.x;

    const bool zero_tile = (t0 < W_WIN);  // block-uniform

    if (!zero_tile) {
        // --- stage 1: closes -> LDS (79 loads, coalesced within waves 0-2) ---
        if (tid < 79) {
            const int t = t0 - W_WIN + tid;
            s_close[tid] = x[((size_t)b * S_LEN + t) * 4 + 3];
        }
        __syncthreads();
        // --- stage 2: 78 shared log-returns ---
        if (tid < 78) {
            s_r[tid] = __logf(s_close[tid + 1] / s_close[tid] + 1e-8f);
        }
        __syncthreads();
        // --- stage 3: per-row window sums + measures (16 lanes, 63 iters) ---
        if (tid < 16) {
            const int m = tid;
            float sum = 0.f, sumsq = 0.f;
            #pragma unroll 7
            for (int j = 0; j < 63; ++j) {       // r indices (t-64)..(t-2)
                const float rv = s_r[m + j];
                sum   += rv;
                sumsq  = fmaf(rv, rv, sumsq);
            }
            const int t = t0 + m;
            const float hi = x[((size_t)b * S_LEN + t) * 4 + 1];
            const float lo = x[((size_t)b * S_LEN + t) * 4 + 2];
            const float log_hl = __logf(hi / lo + 1e-8f);

            // std with ddof=1 over n=63
            const float var = (sumsq - sum * sum * (1.0f / 63.0f)) * (1.0f / 62.0f);
            const float stddev = sqrtf(fmaxf(var, 0.f));         // nan_to_num
            const float mean_r2 = sumsq * (1.0f / 63.0f);
            const float c = 2.0f * LN2F - 1.0f;
            const float gk_arg = 0.5f * log_hl * log_hl - c * mean_r2;
            const float gk = sqrtf(fmaxf(gk_arg, 0.f));          // nan_to_num
            const float pk = sqrtf(log_hl * log_hl * (1.0f / (4.0f * LN2F)));

            s_A[m][0] = stddev;
            s_A[m][1] = gk;
            s_A[m][2] = pk;
            s_A[m][3] = 1.0f;                                    // bias lane
        }
        __syncthreads();
    } else {
        if (tid < 16) {
            s_A[tid][0] = 0.f; s_A[tid][1] = 0.f;
            s_A[tid][2] = 0.f; s_A[tid][3] = 1.0f;
        }
        __syncthreads();
    }

    // --- stage 4: 3->256 projection on the matrix pipe ---
    const int wave = tid >> 5;          // 0..15 -> channel tile
    const int lane = tid & 31;
    const int m    = lane & 15;         // A row / D column index
    const bool hiK = (lane >= 16);      // lanes 16-31 carry K=2,3

    // A fragment: VGPR0 = K (0|2), VGPR1 = K+1 (1|3), rows striped on lanes 0-15/16-31
    v2f Af;
    Af.x = hiK ? s_A[m][2] : s_A[m][0];
    Af.y = hiK ? s_A[m][3] : s_A[m][1];

    // B fragment: B[k][n] with n = lane&15, same K split across lane halves.
    // Row 3 of B is the bias. Use value-selects so EXEC stays all-ones.
    const int h  = wave * 16 + m;       // output channel for this lane's column
    const float w0 = lin_w[h * 3 + 0];
    const float w1 = lin_w[h * 3 + 1];
    const float w2 = lin_w[h * 3 + 2];
    const float bb = lin_b[h];
    v2f Bf;
    Bf.x = hiK ? w2 : w0;
    Bf.y = hiK ? bb : w1;

    v8f Cf = {};
    // (neg_a, A, neg_b, B, c_mod, C, reuse_a, reuse_b)
    Cf = __builtin_amdgcn_wmma_f32_16x16x4_f32(
        false, Af, false, Bf, (short)0, Cf, false, false);

    // --- stage 5: ReLU + store. D layout: VGPR v -> row v + (lane<16?0:8), col lane&15.
    const size_t base = (size_t)b * S_LEN + t0;
    const int row_off = hiK ? 8 : 0;
    #pragma unroll
    for (int v = 0; v < 8; ++v) {
        const int row = v + row_off;
        out[(base + row) * H_DIM + h] = fmaxf(Cf[v], 0.f);
    }
}

extern "C" void kernel_launch(void* const* d_in, const int* in_sizes, int n_in,
                              void* d_out, int out_size, void* d_ws, size_t ws_size,
                              hipStream_t stream) {
    const float* x     = (const float*)d_in[0];   // (32, 8192, 4) f32
    const float* lin_w = (const float*)d_in[1];   // (256, 3) f32
    const float* lin_b = (const float*)d_in[2];   // (256,) f32
    // d_in[3] = window_size (== 64, hardcoded as W_WIN)
    float* out = (float*)d_out;                   // (32, 8192, 256) f32

    dim3 grid(S_LEN / 16, 32);   // 512 time tiles x 32 batch rows
    dim3 block(512);             // 16 wave32 waves (one 16x16 WMMA tile each)
    VolatilityTrackingLayer_kernel<<<grid, block, 0, stream>>>(x, lin_w, lin_b, out);
}